// ConvCRF2d_26233660244510
// MI455X (gfx1250) — compile-verified
//
#include <hip/hip_runtime.h>
#include <cstdint>

// ---------------- problem constants (from reference) ----------------
#define Hh   256
#define Ww   256
#define NPIX (Hh * Ww)
#define NC   21          // classes
#define KK   7           // CRF kernel size
#define SP   3           // KK/2
#define NTAP (KK * KK)   // 49
#define MOM  0.8f
#define EPSC 1e-20f

// conv-iteration tiling
#define TX   16
#define TY   16
#define HX   (TX + 2 * SP)   // 22
#define HY   (TY + 2 * SP)   // 22
#define LDSN (HX * HY)       // 484

// ---------------- CDNA5 async global->LDS helpers ----------------
#define AS1 __attribute__((address_space(1)))
#define AS3 __attribute__((address_space(3)))

__device__ __forceinline__ void async_ld_f32(const float* gsrc, float* ldst) {
#if __has_builtin(__builtin_amdgcn_global_load_async_to_lds_b32)
  __builtin_amdgcn_global_load_async_to_lds_b32(
      (AS1 int*)(uintptr_t)gsrc,
      (AS3 int*)(uint32_t)(uintptr_t)ldst,
      /*imm offset*/ 0, /*cpol*/ 0);
#else
  uint32_t laddr = (uint32_t)(uintptr_t)ldst;   // low 32 bits of flat LDS addr == DS offset
  asm volatile("global_load_async_to_lds_b32 %0, %1, off"
               :: "v"(laddr), "v"(gsrc)
               : "memory");
#endif
}

__device__ __forceinline__ void wait_async0() {
#if __has_builtin(__builtin_amdgcn_s_wait_asynccnt)
  __builtin_amdgcn_s_wait_asynccnt(0);
#else
  asm volatile("s_wait_asynccnt 0" ::: "memory");
#endif
}

// ---------------- halo tile helpers ----------------
// OOB halo slots are written (zeros) exactly once in the prologue; async loads
// only ever target in-bounds slots, so the zeros persist across all channels.
__device__ __forceinline__ void prime_oob_elem(float (*buf)[HX],
                                               int i0, int j0, int e) {
  const int hy = e / HX;
  const int hx = e - hy * HX;
  const int gi = i0 - SP + hy;
  const int gj = j0 - SP + hx;
  if (!((unsigned)gi < (unsigned)Hh && (unsigned)gj < (unsigned)Ww))
    buf[hy][hx] = 0.0f;   // zero-pad (reference unfold pads with zeros)
}

__device__ __forceinline__ void prime_oob(float (*buf)[HX],
                                          int i0, int j0, int tid) {
  prime_oob_elem(buf, i0, j0, tid);
  const int e1 = tid + TX * TY;
  if (e1 < LDSN) prime_oob_elem(buf, i0, j0, e1);
}

__device__ __forceinline__ void load_halo_elem(const float* __restrict__ src,
                                               float (*buf)[HX],
                                               int i0, int j0, int e) {
  const int hy = e / HX;
  const int hx = e - hy * HX;
  const int gi = i0 - SP + hy;
  const int gj = j0 - SP + hx;
  if ((unsigned)gi < (unsigned)Hh && (unsigned)gj < (unsigned)Ww)
    async_ld_f32(src + gi * Ww + gj, &buf[hy][hx]);
}

__device__ __forceinline__ void load_tile(const float* __restrict__ src,
                                          float (*buf)[HX],
                                          int i0, int j0, int tid) {
  load_halo_elem(src, buf, i0, j0, tid);
  const int e1 = tid + TX * TY;
  if (e1 < LDSN) load_halo_elem(src, buf, i0, j0, e1);
}

// ---------------- Kernel 3 (placed first so disasm snippet shows it):
// one CRF mean-field iteration
// out[c,p] = MOM*pred[c,p] + (1-MOM) * sum_k W[k,p]*pred[c, p+off_k]
__global__ __launch_bounds__(256) void crf_iter_kernel(
    const float* __restrict__ pin, const float* __restrict__ Wk,
    float* __restrict__ pout) {
  __shared__ float tile[2][HY][HX];

  const int tx = threadIdx.x, ty = threadIdx.y;
  const int i0 = blockIdx.y * TY, j0 = blockIdx.x * TX;
  const int i = i0 + ty, j = j0 + tx;
  const int p = i * Ww + j;
  const int tid = ty * TX + tx;

  // kick off channel-0 halo tile via the async engine FIRST so the LDS fill
  // overlaps the 49 W vector loads below
  load_tile(pin, tile[0], i0, j0, tid);
  prime_oob(tile[0], i0, j0, tid);
  prime_oob(tile[1], i0, j0, tid);

  // per-pixel 49-tap fused kernel lives in registers across all channels
  float w[NTAP];
#pragma unroll
  for (int k = 0; k < NTAP; ++k) w[k] = Wk[k * NPIX + p];

  wait_async0();
  __syncthreads();

  int cur = 0;
  for (int c = 0; c < NC; ++c) {
    // prefetch next channel's halo tile into the other buffer (overlaps compute)
    if (c + 1 < NC) load_tile(pin + (c + 1) * NPIX, tile[cur ^ 1], i0, j0, tid);

    float msg = 0.f;
#pragma unroll
    for (int kx = 0; kx < KK; ++kx) {
#pragma unroll
      for (int ky = 0; ky < KK; ++ky) {
        msg = __builtin_fmaf(w[kx * KK + ky], tile[cur][ty + kx][tx + ky], msg);
      }
    }
    const float center = tile[cur][ty + SP][tx + SP];
    pout[c * NPIX + p] = MOM * center + (1.0f - MOM) * msg;

    wait_async0();     // next-channel async loads complete
    __syncthreads();   // all waves done reading current buffer / see new buffer
    cur ^= 1;
  }
}

// ---------------- Kernel 1: fused CRF kernel W[k][p] ----------------
// W[k,p] = pw * g_pos[k,p] * rsqrt(sum_k g_pos + eps)
//        + cw * g_com[k,p] * rsqrt(sum_k g_com + eps)
// g_pos = exp(-0.5*(tp0*dx^2 + tp1*dy^2))          (0 if neighbor OOB)
// g_com = exp(-0.5*(tc0*dx^2 + tc1*dy^2 + sum_m tc[2+m]*(img_m[n]-img_m[p])^2))
__global__ __launch_bounds__(256) void build_w_kernel(
    const float* __restrict__ img,
    const float* __restrict__ pos_theta, const float* __restrict__ com_theta,
    const float* __restrict__ pos_w, const float* __restrict__ com_w,
    float* __restrict__ Wk) {
  const int p = blockIdx.x * 256 + threadIdx.x;
  const int i = p >> 8;       // row
  const int j = p & 255;      // col

  const float tp0 = pos_theta[0], tp1 = pos_theta[1];
  const float tc0 = com_theta[0], tc1 = com_theta[1];
  const float tc2 = com_theta[2], tc3 = com_theta[3], tc4 = com_theta[4];

  const float c0 = img[p];
  const float c1 = img[NPIX + p];
  const float c2 = img[2 * NPIX + p];

  float gp[NTAP], gc[NTAP];
  float sp = 0.f, sc = 0.f;

#pragma unroll
  for (int dx = -SP; dx <= SP; ++dx) {
#pragma unroll
    for (int dy = -SP; dy <= SP; ++dy) {
      const int k = (dx + SP) * KK + (dy + SP);
      const int ii = i + dx, jj = j + dy;
      float g1 = 0.f, g2 = 0.f;
      if ((unsigned)ii < (unsigned)Hh && (unsigned)jj < (unsigned)Ww) {
        const float geo = tp0 * (float)(dx * dx) + tp1 * (float)(dy * dy);
        g1 = expf(-0.5f * geo);
        const int q = ii * Ww + jj;
        const float d0 = img[q] - c0;
        const float d1 = img[NPIX + q] - c1;
        const float d2 = img[2 * NPIX + q] - c2;
        const float e = tc0 * (float)(dx * dx) + tc1 * (float)(dy * dy)
                      + tc2 * d0 * d0 + tc3 * d1 * d1 + tc4 * d2 * d2;
        g2 = expf(-0.5f * e);
      }
      gp[k] = g1; gc[k] = g2;
      sp += g1; sc += g2;
    }
  }

  const float a = pos_w[0] * (1.0f / sqrtf(sp + EPSC));
  const float b = com_w[0] * (1.0f / sqrtf(sc + EPSC));
#pragma unroll
  for (int k = 0; k < NTAP; ++k)
    Wk[k * NPIX + p] = a * gp[k] + b * gc[k];
}

// ---------------- Kernel 2/4: log_softmax over 21 channels ----------------
__global__ __launch_bounds__(256) void log_softmax_kernel(
    const float* __restrict__ in, float* __restrict__ out) {
  const int p = blockIdx.x * 256 + threadIdx.x;
  float v[NC];
#pragma unroll
  for (int c = 0; c < NC; ++c) v[c] = in[c * NPIX + p];
  float m = v[0];
#pragma unroll
  for (int c = 1; c < NC; ++c) m = fmaxf(m, v[c]);
  float s = 0.f;
#pragma unroll
  for (int c = 0; c < NC; ++c) s += expf(v[c] - m);
  const float lse = m + logf(s);
#pragma unroll
  for (int c = 0; c < NC; ++c) out[c * NPIX + p] = v[c] - lse;
}

// ---------------- host-side launch ----------------
extern "C" void kernel_launch(void* const* d_in, const int* in_sizes, int n_in,
                              void* d_out, int out_size, void* d_ws, size_t ws_size,
                              hipStream_t stream) {
  const float* image     = (const float*)d_in[0];   // (1,3,256,256)
  const float* unary     = (const float*)d_in[1];   // (1,21,256,256)
  const float* pos_theta = (const float*)d_in[2];   // (1,2,1,1)
  const float* com_theta = (const float*)d_in[3];   // (1,5,1,1)
  const float* pos_w     = (const float*)d_in[4];   // (1,)
  const float* com_w     = (const float*)d_in[5];   // (1,)
  float* out = (float*)d_out;                       // (1,21,256,256)

  float* Wk    = (float*)d_ws;              // 49 * 65536 floats  (12.8 MB)
  float* predA = Wk + NTAP * NPIX;          // 21 * 65536 floats  ( 5.5 MB)
  float* predB = predA + NC * NPIX;         // 21 * 65536 floats  ( 5.5 MB)

  build_w_kernel<<<NPIX / 256, 256, 0, stream>>>(image, pos_theta, com_theta,
                                                 pos_w, com_w, Wk);
  log_softmax_kernel<<<NPIX / 256, 256, 0, stream>>>(unary, predA);

  float* pin = predA;
  float* pout = predB;
  for (int it = 0; it < 5; ++it) {
    crf_iter_kernel<<<dim3(Ww / TX, Hh / TY), dim3(TX, TY), 0, stream>>>(pin, Wk, pout);
    float* t = pin; pin = pout; pout = t;
  }

  log_softmax_kernel<<<NPIX / 256, 256, 0, stream>>>(pin, out);
}